// gnn_basic_72911364817692
// MI455X (gfx1250) — compile-verified
//
#include <hip/hip_runtime.h>

#define N_NODES 100000
#define N_EDGES 1600000
#define KHOPS 5
#define F_IN 64
#define F_HID 128
#define F_OUT 64

typedef float v2f __attribute__((ext_vector_type(2)));
typedef float v8f __attribute__((ext_vector_type(8)));

__device__ __forceinline__ void atomic_add_f32(float* p, float v) {
    // Lowers to global_atomic_add_f32 (device scope, relaxed) on gfx1250.
    __hip_atomic_fetch_add(p, v, __ATOMIC_RELAXED, __HIP_MEMORY_SCOPE_AGENT);
}

// ---------------------------------------------------------------- utilities
__global__ void zero_kernel(float* __restrict__ p, long long n) {
    long long i = blockIdx.x * (long long)blockDim.x + threadIdx.x;
    if (i < n) p[i] = 0.0f;
}

__global__ void deg_kernel(const int* __restrict__ col, const float* __restrict__ ew,
                           float* __restrict__ deg, int E) {
    int e = blockIdx.x * blockDim.x + threadIdx.x;
    if (e < E) atomic_add_f32(&deg[col[e]], ew[e]);
}

__global__ void dis_kernel(float* __restrict__ deg, int n) {
    int i = blockIdx.x * blockDim.x + threadIdx.x;
    if (i < n) {
        float d = deg[i];
        deg[i] = (d > 0.0f) ? rsqrtf(d) : 0.0f;  // deg^-1/2, 0 for isolated
    }
}

__global__ void norm_kernel(const int* __restrict__ row, const int* __restrict__ col,
                            const float* __restrict__ ew, const float* __restrict__ dis,
                            float* __restrict__ nrm, int E) {
    int e = blockIdx.x * blockDim.x + threadIdx.x;
    if (e < E) nrm[e] = dis[row[e]] * ew[e] * dis[col[e]];
}

// ------------------------------------------------------- SpMM (scatter-add)
// One thread per (edge, feature). Consecutive threads share an edge ->
// row/col/norm are wave-uniform loads, gather & atomic stores are coalesced.
template <int LOGF>
__global__ void spmm_kernel(const float* __restrict__ Hin, float* __restrict__ Hout,
                            const int* __restrict__ row, const int* __restrict__ col,
                            const float* __restrict__ nrm, int E) {
    long long gid = blockIdx.x * (long long)blockDim.x + threadIdx.x;
    if (gid >= ((long long)E << LOGF)) return;
    int f = (int)(gid & ((1 << LOGF) - 1));
    int e = (int)(gid >> LOGF);
    float v = nrm[e] * Hin[((size_t)row[e] << LOGF) + f];
    atomic_add_f32(&Hout[((size_t)col[e] << LOGF) + f], v);
}

// ------------------------------------------------------- dense GEMM (WMMA)
// OUT[N x FOUT] (+)= H[N x FIN] @ W[FIN x FOUT], fp32 V_WMMA_F32_16X16X4_F32.
// One wave computes TWO 16-row m-tiles x ALL (FOUT/16) n-tiles:
//  - A fragments reused across every n-tile (H read exactly once)
//  - each B fragment feeds 2 WMMAs
//  - 2*(FOUT/16) independent accumulation chains per wave
template <int FIN, int FOUT, bool ACC>
__global__ void __launch_bounds__(128)
gemm_wmma(const float* __restrict__ H, const float* __restrict__ W,
          float* __restrict__ OUT) {
    constexpr int TILES_N = FOUT / 16;
    constexpr int M_PAIRS = (N_NODES / 16) / 2;  // 3125 (N_NODES % 32 == 0)
    const int wave = threadIdx.x >> 5;
    const int lane = threadIdx.x & 31;
    int pair = blockIdx.x * 4 + wave;
    if (pair >= M_PAIRS) return;

    const int m0 = pair * 32;     // rows m0 .. m0+31
    const int r = lane & 15;      // row (A) / col (B,C) within tile
    const int half = lane >> 4;   // K-pair select for A/B, M-half for C/D

    // A 16x4 layout: lane r, VGPR v -> A[m + r, kk + 2*half + v]
    const float* arow0 = H + (size_t)(m0 + r) * FIN + half * 2;
    const float* arow1 = arow0 + (size_t)16 * FIN;
    // B 4x16 layout: lane r, VGPR v -> B[kk + 2*half + v, n0 + r]
    const float* bbase = W + (size_t)(half * 2) * FOUT + r;
    // C/D 16x16 layout: VGPR v -> row m + v + 8*half, col n0 + r
    float* obase = OUT + (size_t)(m0 + half * 8) * FOUT + r;

    v8f c0[TILES_N], c1[TILES_N];
    const v8f vzero = {};
#pragma unroll
    for (int nt = 0; nt < TILES_N; ++nt) {
        if (ACC) {
            const float* o0 = obase + nt * 16;
            const float* o1 = o0 + (size_t)16 * FOUT;
#pragma unroll
            for (int vi = 0; vi < 8; ++vi) {
                c0[nt][vi] = o0[(size_t)vi * FOUT];
                c1[nt][vi] = o1[(size_t)vi * FOUT];
            }
        } else {
            c0[nt] = vzero;
            c1[nt] = vzero;
        }
    }

#pragma unroll
    for (int kk = 0; kk < FIN; kk += 4) {
        v2f a0, a1;
        a0.x = arow0[kk];
        a0.y = arow0[kk + 1];
        a1.x = arow1[kk];
        a1.y = arow1[kk + 1];
#pragma unroll
        for (int nt = 0; nt < TILES_N; ++nt) {
            v2f b;
            b.x = bbase[(size_t)kk * FOUT + nt * 16];
            b.y = bbase[(size_t)(kk + 1) * FOUT + nt * 16];
            c0[nt] = __builtin_amdgcn_wmma_f32_16x16x4_f32(
                         false, a0, false, b, (short)0, c0[nt], false, false);
            c1[nt] = __builtin_amdgcn_wmma_f32_16x16x4_f32(
                         false, a1, false, b, (short)0, c1[nt], false, false);
        }
    }

#pragma unroll
    for (int nt = 0; nt < TILES_N; ++nt) {
        float* o0 = obase + nt * 16;
        float* o1 = o0 + (size_t)16 * FOUT;
#pragma unroll
        for (int vi = 0; vi < 8; ++vi) {
            o0[(size_t)vi * FOUT] = c0[nt][vi];
            o1[(size_t)vi * FOUT] = c1[nt][vi];
        }
    }
}

// ------------------------------------------------------ bias + leaky ReLU
template <int F>
__global__ void bias_lrelu_kernel(float* __restrict__ out, const float* __restrict__ b,
                                  long long n) {
    long long i = blockIdx.x * (long long)blockDim.x + threadIdx.x;
    if (i < n) {
        float v = out[i] + b[i & (F - 1)];
        out[i] = (v >= 0.0f) ? v : 0.01f * v;
    }
}

static inline unsigned cdiv(long long a, long long b) { return (unsigned)((a + b - 1) / b); }

extern "C" void kernel_launch(void* const* d_in, const int* in_sizes, int n_in,
                              void* d_out, int out_size, void* d_ws, size_t ws_size,
                              hipStream_t stream) {
    const float* x  = (const float*)d_in[0];
    const int*   ei = (const int*)d_in[1];
    const float* ew = (const float*)d_in[2];
    const float* W0 = (const float*)d_in[3];
    const float* b0 = (const float*)d_in[4];
    const float* W1 = (const float*)d_in[5];
    const float* b1 = (const float*)d_in[6];
    float* out = (float*)d_out;

    const int* row = ei;             // edge_index[0]
    const int* col = ei + N_EDGES;   // edge_index[1]

    // Workspace layout (floats): deg | norm | out0 (N*128) | tmp (N*128)
    float* ws   = (float*)d_ws;
    float* deg  = ws;                                     // N_NODES
    float* nrm  = ws + 102400;                            // N_EDGES
    float* out0 = nrm + 1605632;                          // N*128
    float* tmp  = out0 + (size_t)N_NODES * F_HID;         // N*128
    float* bufA = tmp;                                    // N*64 (layer-0 ping)
    float* bufB = tmp + (size_t)N_NODES * F_IN;           // N*64 (layer-0 pong)

    const unsigned gemm_blocks = cdiv((N_NODES / 16) / 2, 4);  // wave = 2 m-tiles

    // ---- GCN symmetric normalization
    zero_kernel<<<cdiv(N_NODES, 256), 256, 0, stream>>>(deg, N_NODES);
    deg_kernel<<<cdiv(N_EDGES, 256), 256, 0, stream>>>(col, ew, deg, N_EDGES);
    dis_kernel<<<cdiv(N_NODES, 256), 256, 0, stream>>>(deg, N_NODES);
    norm_kernel<<<cdiv(N_EDGES, 256), 256, 0, stream>>>(row, col, ew, deg, nrm, N_EDGES);

    // ---- Layer 0: F_IN -> F_HID
    {
        gemm_wmma<F_IN, F_HID, false><<<gemm_blocks, 128, 0, stream>>>(x, W0, out0);
        const float* cur = x;
        for (int k = 1; k <= KHOPS; ++k) {
            float* nxt = (k & 1) ? bufA : bufB;
            long long nf = (long long)N_NODES * F_IN;
            zero_kernel<<<cdiv(nf, 256), 256, 0, stream>>>(nxt, nf);
            long long tot = (long long)N_EDGES << 6;
            spmm_kernel<6><<<cdiv(tot, 256), 256, 0, stream>>>(cur, nxt, row, col, nrm, N_EDGES);
            gemm_wmma<F_IN, F_HID, true><<<gemm_blocks, 128, 0, stream>>>(
                nxt, W0 + (size_t)k * F_IN * F_HID, out0);
            cur = nxt;
        }
        long long nh = (long long)N_NODES * F_HID;
        bias_lrelu_kernel<F_HID><<<cdiv(nh, 256), 256, 0, stream>>>(out0, b0, nh);
    }

    // ---- Layer 1: F_HID -> F_OUT (h0 = out0; ping-pong out0 <-> tmp)
    {
        gemm_wmma<F_HID, F_OUT, false><<<gemm_blocks, 128, 0, stream>>>(out0, W1, out);
        const float* cur = out0;
        for (int k = 1; k <= KHOPS; ++k) {
            float* nxt = (k & 1) ? tmp : out0;  // out0 (h0) free after hop 1
            long long nf = (long long)N_NODES * F_HID;
            zero_kernel<<<cdiv(nf, 256), 256, 0, stream>>>(nxt, nf);
            long long tot = (long long)N_EDGES << 7;
            spmm_kernel<7><<<cdiv(tot, 256), 256, 0, stream>>>(cur, nxt, row, col, nrm, N_EDGES);
            gemm_wmma<F_HID, F_OUT, true><<<gemm_blocks, 128, 0, stream>>>(
                nxt, W1 + (size_t)k * F_HID * F_OUT, out);
            cur = nxt;
        }
        long long no = (long long)N_NODES * F_OUT;
        bias_lrelu_kernel<F_OUT><<<cdiv(no, 256), 256, 0, stream>>>(out, b1, no);
    }
}